// MambaBlock_73143293050997
// MI455X (gfx1250) — compile-verified
//
#include <hip/hip_runtime.h>

// ---------------------------------------------------------------------------
// Mamba block forward for MI455X (gfx1250), wave32.
// GEMMs use V_WMMA_F32_16X16X4_F32 (full fp32 precision), register
// double-buffered LDS staging, conflict-free LDS layouts with K-pair
// interleaved B so fragments load as single ds_load_b64.
// Selective scan uses a 3-pass chunked scan (diagonal A => chunk decay is
// exp(a * sum(dt)), so per-chunk state is 16 floats + one scalar).
// ---------------------------------------------------------------------------

typedef float v2f __attribute__((ext_vector_type(2)));
typedef float v8f __attribute__((ext_vector_type(8)));

#define D_MODEL 768
#define D_STATE 16
#define D_CONV  4
#define D_INNER 1536
#define BATCH   4
#define SEQLEN  4096
#define NROWS   (BATCH * SEQLEN)      // 16384
#define NCHUNK  64
#define CHUNKLEN 64

// ---------------------------------------------------------------------------
// Generic C = A(MxK, row-major) * W(NxK, row-major)^T  with f32 WMMA.
// Block: 256 threads (8 waves), tile 64(M) x 128(N), K-tile 16.
// Wave (wm in 0..3, wn in 0..1) computes a 16x64 strip (4 16x16 accumulators).
// A in LDS at row stride 20 (conflict-free b64 fragment reads).
// B in LDS K-pair interleaved: Bs[(k>>1)*288 + n*2 + (k&1)] so a lane's
// fragment {K, K+1} is one aligned b64; stride 288 = 32 (mod 64) separates
// the two lane halves into disjoint bank ranges.
// ---------------------------------------------------------------------------
__global__ __launch_bounds__(256) void gemm_xWT_wmma(
    float* __restrict__ C, const float* __restrict__ A,
    const float* __restrict__ W, int M, int N, int K)
{
    __shared__ float As[64 * 20];    // 64 rows x 16 K, stride 20
    __shared__ float Bs[8 * 288];    // 8 K-pair rows x 128 n x 2

    const int tid  = threadIdx.x;
    const int lane = tid & 31;
    const int hi   = lane >> 4;      // lane half selects K pair
    const int lm   = lane & 15;
    const int wid  = tid >> 5;
    const int wm   = wid & 3;        // M sub-tile 0..3
    const int wn   = wid >> 2;       // N half 0..1
    const int bm   = blockIdx.x * 64;
    const int bn   = blockIdx.y * 128;

    v8f acc0 = {}, acc1 = {}, acc2 = {}, acc3 = {};

    const int arow = tid >> 2;         // 0..63
    const int acol = (tid & 3) * 4;    // 0,4,8,12
    const int bnt  = tid & 127;        // B staging n (wave covers 32 consec n)
    const int bkh  = (tid >> 7) * 8;   // 0 or 8
    const int brow = bkh >> 1;         // 0 or 4

    const float* __restrict__ Arow = &A[(size_t)(bm + arow) * K + acol];
    const float* __restrict__ Wrow = &W[(size_t)(bn + bnt) * K + bkh];

    // ---- prologue: stage K-tile 0 ----
    float4 ra  = *(const float4*)&Arow[0];
    float4 rb0 = *(const float4*)&Wrow[0];
    float4 rb1 = *(const float4*)&Wrow[4];
    *(float4*)&As[arow * 20 + acol] = ra;
    *(v2f*)&Bs[(brow + 0) * 288 + bnt * 2] = (v2f){rb0.x, rb0.y};
    *(v2f*)&Bs[(brow + 1) * 288 + bnt * 2] = (v2f){rb0.z, rb0.w};
    *(v2f*)&Bs[(brow + 2) * 288 + bnt * 2] = (v2f){rb1.x, rb1.y};
    *(v2f*)&Bs[(brow + 3) * 288 + bnt * 2] = (v2f){rb1.z, rb1.w};

    for (int k0 = 0; k0 < K; k0 += 16) {
        __syncthreads();
        const bool more = (k0 + 16) < K;   // uniform across block
        if (more) {
            if (k0 + 32 < K) {
                __builtin_prefetch(&Arow[k0 + 32], 0, 0);
            }
            ra  = *(const float4*)&Arow[k0 + 16];
            rb0 = *(const float4*)&Wrow[k0 + 16];
            rb1 = *(const float4*)&Wrow[k0 + 20];
        }

#pragma unroll
        for (int kk = 0; kk < 16; kk += 4) {
            // A fragment: lane half hi holds K = kk+2*hi (v0), +1 (v1)
            v2f a2 = *(const v2f*)&As[(wm * 16 + lm) * 20 + kk + 2 * hi];
            const int bro = ((kk >> 1) + hi) * 288 + (wn * 64 + lm) * 2;
            v2f b2;
            b2 = *(const v2f*)&Bs[bro + 0];
            acc0 = __builtin_amdgcn_wmma_f32_16x16x4_f32(false, a2, false, b2, (short)0, acc0, false, false);
            b2 = *(const v2f*)&Bs[bro + 32];
            acc1 = __builtin_amdgcn_wmma_f32_16x16x4_f32(false, a2, false, b2, (short)0, acc1, false, false);
            b2 = *(const v2f*)&Bs[bro + 64];
            acc2 = __builtin_amdgcn_wmma_f32_16x16x4_f32(false, a2, false, b2, (short)0, acc2, false, false);
            b2 = *(const v2f*)&Bs[bro + 96];
            acc3 = __builtin_amdgcn_wmma_f32_16x16x4_f32(false, a2, false, b2, (short)0, acc3, false, false);
        }
        __syncthreads();

        if (more) {
            *(float4*)&As[arow * 20 + acol] = ra;
            *(v2f*)&Bs[(brow + 0) * 288 + bnt * 2] = (v2f){rb0.x, rb0.y};
            *(v2f*)&Bs[(brow + 1) * 288 + bnt * 2] = (v2f){rb0.z, rb0.w};
            *(v2f*)&Bs[(brow + 2) * 288 + bnt * 2] = (v2f){rb1.x, rb1.y};
            *(v2f*)&Bs[(brow + 3) * 288 + bnt * 2] = (v2f){rb1.z, rb1.w};
        }
    }

    // C/D layout: VGPR r -> M = r + 8*hi, N = lm (per 16x16 tile)
    const int crow0 = bm + wm * 16 + hi * 8;
    const int ccol  = bn + wn * 64 + lm;
#pragma unroll
    for (int r = 0; r < 8; ++r) {
        float* cp = &C[(size_t)(crow0 + r) * N + ccol];
        cp[0]  = acc0[r];
        cp[16] = acc1[r];
        cp[32] = acc2[r];
        cp[48] = acc3[r];
    }
}

// ---------------------------------------------------------------------------
// Depthwise causal conv (4 taps) + SiLU. Reads xs-half of xz, writes xs.
// ---------------------------------------------------------------------------
__global__ __launch_bounds__(256) void conv_silu_k(
    float* __restrict__ xs, const float* __restrict__ xz,
    const float* __restrict__ wconv)
{
    const int idx = blockIdx.x * 256 + threadIdx.x;   // NROWS * D_INNER
    const int d = idx % D_INNER;
    const int l = (idx / D_INNER) % SEQLEN;
    const int b = idx / (D_INNER * SEQLEN);
    const float w0 = wconv[d * 4 + 0];
    const float w1 = wconv[d * 4 + 1];
    const float w2 = wconv[d * 4 + 2];
    const float w3 = wconv[d * 4 + 3];
    const size_t base = (size_t)(b * SEQLEN) * (2 * D_INNER) + d;
    float acc = w3 * xz[base + (size_t)l * (2 * D_INNER)];
    if (l >= 1) acc += w2 * xz[base + (size_t)(l - 1) * (2 * D_INNER)];
    if (l >= 2) acc += w1 * xz[base + (size_t)(l - 2) * (2 * D_INNER)];
    if (l >= 3) acc += w0 * xz[base + (size_t)(l - 3) * (2 * D_INNER)];
    const float sig = 1.0f / (1.0f + __expf(-acc));
    xs[(size_t)(b * SEQLEN + l) * D_INNER + d] = acc * sig;
}

// ---------------------------------------------------------------------------
// x-proj: proj[row][j] = dot(xs[row], w_xproj[j]), j = 0..32 (N=33, too small
// for WMMA; 1.7 GFLOP total). One row per block; w_xproj (200 KB) is L2-hot.
// ---------------------------------------------------------------------------
__global__ __launch_bounds__(64) void xproj_k(
    float* __restrict__ proj, const float* __restrict__ xs,
    const float* __restrict__ wx)
{
    const int row = blockIdx.x;
    const int j   = threadIdx.x;
    if (j >= 33) return;
    const float4* xr = (const float4*)&xs[(size_t)row * D_INNER];
    const float4* wr = (const float4*)&wx[(size_t)j * D_INNER];
    float acc = 0.0f;
    for (int i = 0; i < D_INNER / 4; ++i) {
        float4 xv = xr[i];
        float4 wv = wr[i];
        acc += xv.x * wv.x + xv.y * wv.y + xv.z * wv.z + xv.w * wv.w;
    }
    proj[(size_t)row * 33 + j] = acc;
}

// ---------------------------------------------------------------------------
// delta[row][d] = softplus(proj[row][0] * w_dt[d] + b_dt[d])
// ---------------------------------------------------------------------------
__global__ __launch_bounds__(256) void delta_k(
    float* __restrict__ delta, const float* __restrict__ proj,
    const float* __restrict__ w_dt, const float* __restrict__ b_dt)
{
    const int idx = blockIdx.x * 256 + threadIdx.x;  // NROWS * D_INNER
    const int d = idx % D_INNER;
    const int row = idx / D_INNER;
    const float x = proj[(size_t)row * 33] * w_dt[d] + b_dt[d];
    const float sp = (x > 20.0f) ? x : log1pf(__expf(x));
    delta[(size_t)row * D_INNER + d] = sp;
}

// ---------------------------------------------------------------------------
// Scan pass A: per-chunk local scan with h0 = 0. Records chunk-final state
// and sum(dt) (chunk decay = exp(a * sum_dt)). Writes y_local + D*x.
// Grid: B * NCHUNK * (D_INNER/256) blocks; one thread per channel d.
// ---------------------------------------------------------------------------
__global__ __launch_bounds__(256) void scan_chunk_local(
    float* __restrict__ yb, float* __restrict__ hloc, float* __restrict__ sdtb,
    const float* __restrict__ delta, const float* __restrict__ xs,
    const float* __restrict__ proj, const float* __restrict__ a_log,
    const float* __restrict__ dpar)
{
    __shared__ float bc[CHUNKLEN * 32];   // [t][n]: n<16 -> B, n>=16 -> C
    const int tid  = threadIdx.x;
    const int dblk = blockIdx.x % (D_INNER / 256);
    const int c    = (blockIdx.x / (D_INNER / 256)) % NCHUNK;
    const int b    = blockIdx.x / ((D_INNER / 256) * NCHUNK);
    const int d    = dblk * 256 + tid;
    const int l0   = c * CHUNKLEN;

#pragma unroll
    for (int i = 0; i < 8; ++i) {
        const int fl = tid * 8 + i;                 // 0..2047
        const int tl = fl >> 5, n = fl & 31;
        bc[fl] = proj[(size_t)(b * SEQLEN + l0 + tl) * 33 + 1 + n];
    }
    __syncthreads();

    float am[D_STATE], h[D_STATE];
#pragma unroll
    for (int n = 0; n < D_STATE; ++n) {
        am[n] = -__expf(a_log[(size_t)d * D_STATE + n]);
        h[n] = 0.0f;
    }
    float sdt = 0.0f;
    const float Dp = dpar[d];

    for (int t = 0; t < CHUNKLEN; ++t) {
        const size_t row = (size_t)(b * SEQLEN + l0 + t);
        const float dt = delta[row * D_INNER + d];
        const float x  = xs[row * D_INNER + d];
        sdt += dt;
        const float dtx = dt * x;
        const float* bcr = &bc[t * 32];
        float y = 0.0f;
#pragma unroll
        for (int n = 0; n < D_STATE; ++n) {
            const float ab = __expf(dt * am[n]);
            h[n] = ab * h[n] + dtx * bcr[n];
            y += bcr[16 + n] * h[n];
        }
        yb[row * D_INNER + d] = y + Dp * x;
    }

    const size_t st = (size_t)(b * NCHUNK + c) * D_INNER + d;
#pragma unroll
    for (int n = 0; n < D_STATE; ++n) hloc[st * D_STATE + n] = h[n];
    sdtb[st] = sdt;
}

// ---------------------------------------------------------------------------
// Scan pass B: sequential combine over chunks (only 64 steps), one thread
// per (b, d, n). hin[c] = state entering chunk c.
// ---------------------------------------------------------------------------
__global__ __launch_bounds__(256) void scan_combine(
    float* __restrict__ hin, const float* __restrict__ hloc,
    const float* __restrict__ sdtb, const float* __restrict__ a_log)
{
    const int idx = blockIdx.x * 256 + threadIdx.x;  // B * D_INNER * 16
    const int n = idx & 15;
    const int d = (idx >> 4) % D_INNER;
    const int b = idx / (16 * D_INNER);
    const float a = -__expf(a_log[(size_t)d * D_STATE + n]);
    float h = 0.0f;
    for (int c = 0; c < NCHUNK; ++c) {
        const size_t st = (size_t)(b * NCHUNK + c) * D_INNER + d;
        hin[st * D_STATE + n] = h;
        h = __expf(a * sdtb[st]) * h + hloc[st * D_STATE + n];
    }
}

// ---------------------------------------------------------------------------
// Scan pass C: y += sum_n C[t][n] * exp(a_n * cumsum(dt)) * h_in[n],
// fused with the output gate: y_final = y * silu(z). Overwrites yb in place.
// ---------------------------------------------------------------------------
__global__ __launch_bounds__(256) void scan_fixup_gate(
    float* __restrict__ yb, const float* __restrict__ hin,
    const float* __restrict__ delta, const float* __restrict__ proj,
    const float* __restrict__ xz, const float* __restrict__ a_log)
{
    __shared__ float cs[CHUNKLEN * 16];
    const int tid  = threadIdx.x;
    const int dblk = blockIdx.x % (D_INNER / 256);
    const int c    = (blockIdx.x / (D_INNER / 256)) % NCHUNK;
    const int b    = blockIdx.x / ((D_INNER / 256) * NCHUNK);
    const int d    = dblk * 256 + tid;
    const int l0   = c * CHUNKLEN;

#pragma unroll
    for (int i = 0; i < 4; ++i) {
        const int fl = tid * 4 + i;                 // 0..1023
        const int tl = fl >> 4, n = fl & 15;
        cs[fl] = proj[(size_t)(b * SEQLEN + l0 + tl) * 33 + 17 + n];
    }
    __syncthreads();

    const size_t st = (size_t)(b * NCHUNK + c) * D_INNER + d;
    float am[D_STATE], hv[D_STATE];
#pragma unroll
    for (int n = 0; n < D_STATE; ++n) {
        am[n] = -__expf(a_log[(size_t)d * D_STATE + n]);
        hv[n] = hin[st * D_STATE + n];
    }

    float S = 0.0f;
    for (int t = 0; t < CHUNKLEN; ++t) {
        const size_t row = (size_t)(b * SEQLEN + l0 + t);
        const float dt = delta[row * D_INNER + d];
        S += dt;
        const float* csr = &cs[t * 16];
        float corr = 0.0f;
#pragma unroll
        for (int n = 0; n < D_STATE; ++n) {
            corr += csr[n] * __expf(am[n] * S) * hv[n];
        }
        const float yv = yb[row * D_INNER + d] + corr;
        const float z  = xz[row * (2 * D_INNER) + D_INNER + d];
        const float sig = 1.0f / (1.0f + __expf(-z));
        yb[row * D_INNER + d] = yv * (z * sig);
    }
}

// ---------------------------------------------------------------------------
extern "C" void kernel_launch(void* const* d_in, const int* in_sizes, int n_in,
                              void* d_out, int out_size, void* d_ws, size_t ws_size,
                              hipStream_t stream)
{
    const float* x      = (const float*)d_in[0];
    const float* w_in   = (const float*)d_in[1];
    const float* w_conv = (const float*)d_in[2];
    const float* w_xp   = (const float*)d_in[3];
    const float* w_dt   = (const float*)d_in[4];
    const float* b_dt   = (const float*)d_in[5];
    const float* a_log  = (const float*)d_in[6];
    const float* d_par  = (const float*)d_in[7];
    const float* w_out  = (const float*)d_in[8];
    float* out = (float*)d_out;

    float* ws    = (float*)d_ws;
    float* xz    = ws;                                        // NROWS*3072
    float* xs    = xz    + (size_t)NROWS * (2 * D_INNER);     // NROWS*1536
    float* delta = xs    + (size_t)NROWS * D_INNER;           // NROWS*1536
    float* proj  = delta + (size_t)NROWS * D_INNER;           // NROWS*33
    float* yb    = proj  + (size_t)NROWS * 33;                // NROWS*1536
    float* hloc  = yb    + (size_t)NROWS * D_INNER;           // B*NC*1536*16
    float* sdtb  = hloc  + (size_t)BATCH * NCHUNK * D_INNER * D_STATE;
    float* hin   = sdtb  + (size_t)BATCH * NCHUNK * D_INNER;  // B*NC*1536*16

    // 1) in-projection: xz = x @ w_in^T   (16384 x 3072, K=768)
    gemm_xWT_wmma<<<dim3(NROWS / 64, (2 * D_INNER) / 128), 256, 0, stream>>>(
        xz, x, w_in, NROWS, 2 * D_INNER, D_MODEL);
    // 2) depthwise conv + SiLU
    conv_silu_k<<<(NROWS * D_INNER) / 256, 256, 0, stream>>>(xs, xz, w_conv);
    // 3) x-proj (delta_raw | B | C)
    xproj_k<<<NROWS, 64, 0, stream>>>(proj, xs, w_xp);
    // 4) delta = softplus(...)
    delta_k<<<(NROWS * D_INNER) / 256, 256, 0, stream>>>(delta, proj, w_dt, b_dt);
    // 5-7) chunked selective scan + gating
    scan_chunk_local<<<BATCH * NCHUNK * (D_INNER / 256), 256, 0, stream>>>(
        yb, hloc, sdtb, delta, xs, proj, a_log, d_par);
    scan_combine<<<(BATCH * D_INNER * D_STATE) / 256, 256, 0, stream>>>(
        hin, hloc, sdtb, a_log);
    scan_fixup_gate<<<BATCH * NCHUNK * (D_INNER / 256), 256, 0, stream>>>(
        yb, hin, delta, proj, xz, a_log);
    // 8) out-projection: out = (y * silu(z)) @ w_out^T  (16384 x 768, K=1536)
    gemm_xWT_wmma<<<dim3(NROWS / 64, D_MODEL / 128), 256, 0, stream>>>(
        out, yb, w_out, NROWS, D_MODEL, D_INNER);
}